// Bipartite_4647154614416
// MI455X (gfx1250) — compile-verified
//
#include <hip/hip_runtime.h>
#include <math.h>

#define T_DIM 1024
#define A_DIM 512
#define D_DIM 128
#define EPS 1e-5f
#define SLOPE 0.01f
#define NPART ((T_DIM/16)*(A_DIM/16))   // 2048 edge blocks

typedef __attribute__((ext_vector_type(2))) float v2f;
typedef __attribute__((ext_vector_type(8))) float v8f;

// ---------------------------------------------------------------------------
// K1: fp32 WMMA projections.
//   Psrc[t,j] = sum_k task[t,k]  * W1[j,     k]   (W1 is [128,256] row-major)
//   Pdst[a,j] = sum_k agent[a,k] * W1[j, 128+k]
// One wave (32 lanes) per 16x16 output tile; K looped in steps of 4 using
// V_WMMA_F32_16X16X4_F32 (A: 16x4, B: 4x16, C/D: 16x16 f32).
// ---------------------------------------------------------------------------
__global__ void k1_proj(const float* __restrict__ task,
                        const float* __restrict__ agent,
                        const float* __restrict__ W1,
                        float* __restrict__ Psrc,
                        float* __restrict__ Pdst) {
  int bid = blockIdx.x;
  int nt = bid & 7;          // 8 column tiles (D=128)
  int mt = bid >> 3;         // 96 row tiles (64 task + 32 agent)

  const float* X;
  float* OUT;
  int wOff, rowBase;
  if (mt < T_DIM / 16) { X = task;  OUT = Psrc; wOff = 0;     rowBase = mt * 16; }
  else                 { X = agent; OUT = Pdst; wOff = D_DIM; rowBase = (mt - T_DIM / 16) * 16; }

  int lane = threadIdx.x;    // 0..31, EXEC all ones (full wave, no divergence)
  int half = lane >> 4;      // selects K pair {0,1} vs {2,3}
  int l    = lane & 15;      // M index for A, N index for B
  int nBase = nt * 16;

  // A row for this lane (M = rowBase + l), B row = W1 row (nBase + l)
  const float* arow = X  + (size_t)(rowBase + l) * D_DIM;
  const float* brow = W1 + (size_t)(nBase + l) * (2 * D_DIM) + wOff;

  v8f acc = {};
  #pragma unroll
  for (int kb = 0; kb < D_DIM; kb += 4) {
    int k = kb + 2 * half;                       // even -> 8B-aligned pair
    float2 ap = ((const float2*)arow)[k >> 1];   // A[M][k], A[M][k+1]
    float2 bp = ((const float2*)brow)[k >> 1];   // B[k][N], B[k+1][N] (= W1[N][k..k+1])
    v2f av; av.x = ap.x; av.y = ap.y;
    v2f bv; bv.x = bp.x; bv.y = bp.y;
    acc = __builtin_amdgcn_wmma_f32_16x16x4_f32(
        /*neg_a=*/false, av, /*neg_b=*/false, bv,
        /*c_mod=*/(short)0, acc, /*reuse_a=*/false, /*reuse_b=*/false);
  }

  // C/D layout: VGPR r -> M = r + 8*half, N = l
  float* orow = OUT + (size_t)(rowBase + 8 * half) * D_DIM + nBase + l;
  #pragma unroll
  for (int r = 0; r < 8; ++r)
    orow[(size_t)r * D_DIM] = acc[r];
}

// ---------------------------------------------------------------------------
// K2: per-channel BN1 stats (factorized over the two axes) + fold affine.
// Single block, 128 threads, thread j owns column j (coalesced column walks).
//   U = alpha * Psrc            (in place)
//   V = alpha * Pdst + c        (in place)
// ---------------------------------------------------------------------------
__global__ void k2_stats(float* __restrict__ Psrc, float* __restrict__ Pdst,
                         const float* __restrict__ gamma1,
                         const float* __restrict__ beta1) {
  int j = threadIdx.x;
  float s = 0.f, q = 0.f;
  for (int t = 0; t < T_DIM; ++t) { float v = Psrc[t * D_DIM + j]; s += v; q += v * v; }
  float ms = s * (1.f / T_DIM);
  float vs = q * (1.f / T_DIM) - ms * ms;

  s = 0.f; q = 0.f;
  for (int a = 0; a < A_DIM; ++a) { float v = Pdst[a * D_DIM + j]; s += v; q += v * v; }
  float md = s * (1.f / A_DIM);
  float vd = q * (1.f / A_DIM) - md * md;

  float mean  = ms + md;
  float var   = vs + vd;                 // exact factorization of edge variance
  float alpha = gamma1[j] * rsqrtf(var + EPS);
  float c     = beta1[j] - mean * alpha;

  for (int t = 0; t < T_DIM; ++t) Psrc[t * D_DIM + j] *= alpha;
  for (int a = 0; a < A_DIM; ++a) Pdst[a * D_DIM + j] = fmaf(Pdst[a * D_DIM + j], alpha, c);
}

// ---------------------------------------------------------------------------
// K3: edge kernel. 16(t) x 16(a) tile per block, LDS-cached U/V rows
// (stride 132 floats to spread the 64 LDS banks), branchless LeakyReLU:
//   lrelu(x) = max(x,0) + slope*min(x,0)
// Emits raw score S[t,a] and deterministic per-block (sum, sumsq) partials.
// ---------------------------------------------------------------------------
#define LDSW 132
__global__ void k3_edges(const float* __restrict__ U, const float* __restrict__ V,
                         const float* __restrict__ W2,
                         float* __restrict__ S, float* __restrict__ partials) {
  __shared__ float Ut[16 * LDSW];
  __shared__ float Va[16 * LDSW];
  __shared__ float w2s[D_DIM];
  __shared__ float redS[256];
  __shared__ float redQ[256];

  int tid = threadIdx.x;
  int tBase = blockIdx.x * 16;
  int aBase = blockIdx.y * 16;

  for (int i = tid; i < 16 * D_DIM; i += 256) {
    int r = i >> 7, c = i & (D_DIM - 1);
    Ut[r * LDSW + c] = U[(size_t)(tBase + r) * D_DIM + c];
    Va[r * LDSW + c] = V[(size_t)(aBase + r) * D_DIM + c];
  }
  if (tid < D_DIM) w2s[tid] = W2[tid];
  __syncthreads();

  int tx = tid & 15;   // a within tile
  int ty = tid >> 4;   // t within tile
  const float* u = Ut + ty * LDSW;
  const float* v = Va + tx * LDSW;

  float acc = 0.f;
  #pragma unroll 8
  for (int j = 0; j < D_DIM; ++j) {
    float x = u[j] + v[j];
    float h = fmaxf(x, 0.f) + SLOPE * fminf(x, 0.f);
    acc = fmaf(w2s[j], h, acc);
  }

  S[(size_t)(tBase + ty) * A_DIM + aBase + tx] = acc;

  redS[tid] = acc;
  redQ[tid] = acc * acc;
  __syncthreads();
  for (int step = 128; step > 0; step >>= 1) {
    if (tid < step) { redS[tid] += redS[tid + step]; redQ[tid] += redQ[tid + step]; }
    __syncthreads();
  }
  if (tid == 0) {
    int b = blockIdx.y * gridDim.x + blockIdx.x;
    partials[2 * b]     = redS[0];
    partials[2 * b + 1] = redQ[0];
  }
}

// ---------------------------------------------------------------------------
// K3b: deterministic reduction of 2048 partials -> (sum, sumsq).
// ---------------------------------------------------------------------------
__global__ void k3b_reduce(const float* __restrict__ partials,
                           float* __restrict__ accum) {
  __shared__ float rs[256], rq[256];
  int tid = threadIdx.x;
  float s = 0.f, q = 0.f;
  for (int i = tid; i < NPART; i += 256) { s += partials[2 * i]; q += partials[2 * i + 1]; }
  rs[tid] = s; rq[tid] = q;
  __syncthreads();
  for (int step = 128; step > 0; step >>= 1) {
    if (tid < step) { rs[tid] += rs[tid + step]; rq[tid] += rq[tid + step]; }
    __syncthreads();
  }
  if (tid == 0) { accum[0] = rs[0]; accum[1] = rq[0]; }
}

// ---------------------------------------------------------------------------
// K4: BN2 normalize + finished mask + transpose to [A, T] output.
// 32x32 LDS tile (pad to 33) so both the S reads and out writes coalesce.
// ---------------------------------------------------------------------------
__global__ void k4_final(const float* __restrict__ S, const float* __restrict__ accum,
                         const float* __restrict__ gamma2, const float* __restrict__ beta2,
                         const int* __restrict__ finished, float* __restrict__ out) {
  __shared__ float tile[32][33];
  int tid = threadIdx.x;
  int tBase = blockIdx.x * 32;
  int aBase = blockIdx.y * 32;

  const float invN = 1.f / (float)(T_DIM * A_DIM);
  float m2 = accum[0] * invN;
  float v2 = accum[1] * invN - m2 * m2;
  float scale = gamma2[0] * rsqrtf(v2 + EPS);
  float bias  = beta2[0];

  for (int i = tid; i < 1024; i += 256) {
    int tt = i >> 5, aa = i & 31;
    tile[tt][aa] = S[(size_t)(tBase + tt) * A_DIM + aBase + aa];  // coalesced in aa
  }
  __syncthreads();
  for (int i = tid; i < 1024; i += 256) {
    int aa = i >> 5, tt = i & 31;
    float sv = fmaf(tile[tt][aa] - m2, scale, bias);
    if (finished[tBase + tt]) sv = -__builtin_inff();
    out[(size_t)(aBase + aa) * T_DIM + tBase + tt] = sv;          // coalesced in tt
  }
}

// ---------------------------------------------------------------------------
extern "C" void kernel_launch(void* const* d_in, const int* in_sizes, int n_in,
                              void* d_out, int out_size, void* d_ws, size_t ws_size,
                              hipStream_t stream) {
  const float* task  = (const float*)d_in[0];   // [1024,128]
  const float* agent = (const float*)d_in[1];   // [512,128]
  const float* W1    = (const float*)d_in[2];   // [128,256]
  const float* g1    = (const float*)d_in[3];   // [128]
  const float* b1    = (const float*)d_in[4];   // [128]
  const float* W2    = (const float*)d_in[5];   // [1,128]
  const float* g2    = (const float*)d_in[6];   // [1]
  const float* b2    = (const float*)d_in[7];   // [1]
  const int*   fin   = (const int*)d_in[8];     // [1024]
  float* out = (float*)d_out;                   // [512,1024]

  float* ws       = (float*)d_ws;
  float* Psrc     = ws;                          // 1024*128
  float* Pdst     = Psrc + T_DIM * D_DIM;        // 512*128
  float* S        = Pdst + A_DIM * D_DIM;        // 1024*512
  float* partials = S + (size_t)T_DIM * A_DIM;   // 2*NPART
  float* accum    = partials + 2 * NPART;        // 2

  k1_proj   <<<dim3((T_DIM/16 + A_DIM/16) * (D_DIM/16)), dim3(32),  0, stream>>>(task, agent, W1, Psrc, Pdst);
  k2_stats  <<<dim3(1),                                  dim3(128), 0, stream>>>(Psrc, Pdst, g1, b1);
  k3_edges  <<<dim3(T_DIM/16, A_DIM/16),                 dim3(256), 0, stream>>>(Psrc, Pdst, W2, S, partials);
  k3b_reduce<<<dim3(1),                                  dim3(256), 0, stream>>>(partials, accum);
  k4_final  <<<dim3(T_DIM/32, A_DIM/32),                 dim3(256), 0, stream>>>(S, accum, g2, b2, fin, out);
}